// OptNet_42365557408293
// MI455X (gfx1250) — compile-verified
//
#include <hip/hip_runtime.h>
#include <math.h>

typedef float v2f __attribute__((ext_vector_type(2)));
typedef float v8f __attribute__((ext_vector_type(8)));

#define NX    729
#define NEQ   324
#define NB    32768
#define QPEN  0.1f
#define NPAD  768          // padded K/N for W
#define YCOLS 736          // padded column count for Y storage (729 cols of A + 1 col for b)

// ---------------------------------------------------------------- setup: M = A A^T
__global__ void k_compute_M(const float* __restrict__ A, float* __restrict__ M) {
    int idx = blockIdx.x * blockDim.x + threadIdx.x;
    if (idx >= NEQ * NEQ) return;
    int i = idx / NEQ, j = idx % NEQ;
    const float* ai = A + (size_t)i * NX;
    const float* aj = A + (size_t)j * NX;
    float s = 0.f;
    for (int k = 0; k < NX; ++k) s += ai[k] * aj[k];
    M[idx] = s;
}

// ---------------------------------------------------------------- setup: in-place Cholesky (lower)
__global__ void k_cholesky(float* __restrict__ M) {
    int tid = threadIdx.x;
    for (int j = 0; j < NEQ; ++j) {
        if (tid == 0) M[j * NEQ + j] = sqrtf(M[j * NEQ + j]);
        __syncthreads();
        float d = M[j * NEQ + j];
        for (int i = j + 1 + tid; i < NEQ; i += blockDim.x) M[i * NEQ + j] /= d;
        __syncthreads();
        for (int i = j + 1 + tid; i < NEQ; i += blockDim.x) {
            float lij = M[i * NEQ + j];
            for (int l = j + 1; l <= i; ++l) M[i * NEQ + l] -= lij * M[l * NEQ + j];
        }
        __syncthreads();
    }
}

// ---------------------------------------------------------------- setup: solve M Y = [A | b]
// warp-per-column forward/backward substitution; z lives in LDS
__global__ void k_solve(const float* __restrict__ L, const float* __restrict__ A,
                        const float* __restrict__ b, float* __restrict__ Y) {
    __shared__ float zsh[8][NEQ];
    int lane = threadIdx.x & 31;
    int warp = threadIdx.x >> 5;
    int col  = blockIdx.x * 8 + warp;
    if (col >= NX + 1) return;              // whole warp exits together
    float* z = zsh[warp];

    // forward: L z = rhs
    for (int i = 0; i < NEQ; ++i) {
        const float* Li = L + (size_t)i * NEQ;
        float partial = 0.f;
        for (int k = lane; k < i; k += 32) partial += Li[k] * z[k];
        for (int off = 16; off > 0; off >>= 1) partial += __shfl_down(partial, off, 32);
        if (lane == 0) {
            float rhs = (col < NX) ? A[(size_t)i * NX + col] : b[i];
            z[i] = (rhs - partial) / Li[i];
        }
        __builtin_amdgcn_wave_barrier();
    }
    // backward: L^T y = z (overwrite z)
    for (int i = NEQ - 1; i >= 0; --i) {
        float partial = 0.f;
        for (int k = i + 1 + lane; k < NEQ; k += 32) partial += L[(size_t)k * NEQ + i] * z[k];
        for (int off = 16; off > 0; off >>= 1) partial += __shfl_down(partial, off, 32);
        if (lane == 0) z[i] = (z[i] - partial) / L[i * NEQ + i];
        __builtin_amdgcn_wave_barrier();
    }
    for (int i = lane; i < NEQ; i += 32) Y[(size_t)i * YCOLS + col] = z[i];
}

// ---------------------------------------------------------------- setup: W = (I - A^T Y)/q (zero-padded), c = A^T y
__global__ void k_build_W(const float* __restrict__ A, const float* __restrict__ Y,
                          float* __restrict__ W, float* __restrict__ cvec) {
    int idx = blockIdx.x * blockDim.x + threadIdx.x;
    if (idx >= NPAD * NPAD) return;
    int r = idx / NPAD;          // k-index of W
    int col = idx % NPAD;
    float w = 0.f;
    if (r < NX && col < NX) {
        float s = 0.f;
        for (int i = 0; i < NEQ; ++i) s += A[(size_t)i * NX + r] * Y[(size_t)i * YCOLS + col];
        w = (((r == col) ? 1.f : 0.f) - s) * (1.0f / QPEN);
    }
    W[idx] = w;
    if (r == 0) {
        float cv = 0.f;
        if (col < NX)
            for (int i = 0; i < NEQ; ++i) cv += A[(size_t)i * NX + col] * Y[(size_t)i * YCOLS + NX];
        cvec[col] = cv;
    }
}

// ---------------------------------------------------------------- main: X = P * W + c via V_WMMA_F32_16X16X4_F32
// block = 256 threads = 8 waves (4 M x 2 N), wave tile 64x64, block tile 256x128
// Branch-free K loop: K=0..724 fully in-bounds; single clamped tail step for k=728..731
// (invalid k values hit zero-padded W rows, so clamped loads contribute exactly 0).
__global__ __launch_bounds__(256) void k_gemm(const float* __restrict__ P,
                                              const float* __restrict__ W,
                                              const float* __restrict__ cvec,
                                              float* __restrict__ X) {
    int lane  = threadIdx.x & 31;
    int wave  = threadIdx.x >> 5;
    int wm    = wave >> 1;
    int wn    = wave & 1;
    int Mbase = blockIdx.x * 256 + wm * 64;
    int Nbase = blockIdx.y * 128 + wn * 64;

    int lhalf = lane >> 4;           // 0: lanes 0-15, 1: lanes 16-31
    int lmod  = lane & 15;

    int rowoff[4];                   // P element offsets, loop-invariant
    int colb[4];                     // W column indices, loop-invariant
#pragma unroll
    for (int mt = 0; mt < 4; ++mt) rowoff[mt] = (Mbase + mt * 16 + lmod) * NX;
#pragma unroll
    for (int nt = 0; nt < 4; ++nt) colb[nt] = Nbase + nt * 16 + lmod;

    v8f acc[4][4];
#pragma unroll
    for (int mt = 0; mt < 4; ++mt)
#pragma unroll
        for (int nt = 0; nt < 4; ++nt) acc[mt][nt] = (v8f){0.f,0.f,0.f,0.f,0.f,0.f,0.f,0.f};

    // ---- main loop: all indices provably < 729 (k0+1 <= 727), zero guards ----
    for (int K = 0; K + 4 <= NX; K += 4) {          // K = 0,4,...,724
        int k0 = K + 2 * lhalf;                     // ISA f32 A/B layout: VGPR0=k0, VGPR1=k0+1
        v2f a[4], bfr[4];
#pragma unroll
        for (int mt = 0; mt < 4; ++mt) {
            a[mt].x = P[rowoff[mt] + k0];
            a[mt].y = P[rowoff[mt] + k0 + 1];
        }
#pragma unroll
        for (int nt = 0; nt < 4; ++nt) {
            bfr[nt].x = W[(size_t)k0 * NPAD + colb[nt]];
            bfr[nt].y = W[(size_t)(k0 + 1) * NPAD + colb[nt]];
        }
#pragma unroll
        for (int mt = 0; mt < 4; ++mt)
#pragma unroll
            for (int nt = 0; nt < 4; ++nt)
                acc[mt][nt] = __builtin_amdgcn_wmma_f32_16x16x4_f32(
                    false, a[mt], false, bfr[nt], (short)0, acc[mt][nt], false, false);
    }

    // ---- tail step: k = 728..731; clamp P index (in-bounds), W rows >=729 are zero ----
    {
        int k0  = 728 + 2 * lhalf;
        int kc0 = (k0     < NX) ? k0     : (NX - 1);
        int kc1 = (k0 + 1 < NX) ? k0 + 1 : (NX - 1);
        v2f a[4], bfr[4];
#pragma unroll
        for (int mt = 0; mt < 4; ++mt) {
            a[mt].x = P[rowoff[mt] + kc0];
            a[mt].y = P[rowoff[mt] + kc1];
        }
#pragma unroll
        for (int nt = 0; nt < 4; ++nt) {
            bfr[nt].x = W[(size_t)k0 * NPAD + colb[nt]];
            bfr[nt].y = W[(size_t)(k0 + 1) * NPAD + colb[nt]];
        }
#pragma unroll
        for (int mt = 0; mt < 4; ++mt)
#pragma unroll
            for (int nt = 0; nt < 4; ++nt)
                acc[mt][nt] = __builtin_amdgcn_wmma_f32_16x16x4_f32(
                    false, a[mt], false, bfr[nt], (short)0, acc[mt][nt], false, false);
    }

    // store: C/D layout — VGPR v: lanes 0-15 -> M=v, lanes 16-31 -> M=v+8; N = lane%16
#pragma unroll
    for (int nt = 0; nt < 4; ++nt) {
        int col = colb[nt];
        if (col < NX) {
            float cv = cvec[col];
#pragma unroll
            for (int mt = 0; mt < 4; ++mt) {
#pragma unroll
                for (int v = 0; v < 8; ++v) {
                    int row = Mbase + mt * 16 + v + lhalf * 8;
                    X[(size_t)row * NX + col] = acc[mt][nt][v] + cv;
                }
            }
        }
    }
}

// ---------------------------------------------------------------- launcher
extern "C" void kernel_launch(void* const* d_in, const int* in_sizes, int n_in,
                              void* d_out, int out_size, void* d_ws, size_t ws_size,
                              hipStream_t stream) {
    const float* puzzles = (const float*)d_in[0];   // [32768,9,9,9] == P (note: p=-P, -p=P)
    const float* A       = (const float*)d_in[1];   // [324,729]
    const float* b       = (const float*)d_in[2];   // [324]
    float*       X       = (float*)d_out;           // [32768,729]

    float* ws   = (float*)d_ws;
    float* M    = ws;                                   // 324*324
    float* Y    = M + NEQ * NEQ;                        // 324*736 (col 729 = M^-1 b)
    float* W    = Y + NEQ * YCOLS;                      // 768*768 zero-padded
    float* cvec = W + NPAD * NPAD;                      // 768

    // 1) M = A A^T
    {
        int total = NEQ * NEQ;
        k_compute_M<<<(total + 255) / 256, 256, 0, stream>>>(A, M);
    }
    // 2) Cholesky in place
    k_cholesky<<<1, 256, 0, stream>>>(M);
    // 3) Y = M^-1 [A | b]  (warp per column)
    k_solve<<<(NX + 1 + 7) / 8, 256, 0, stream>>>(M, A, b, Y);
    // 4) W = (I - A^T Y)/q (padded to 768x768), c = A^T (M^-1 b)
    k_build_W<<<(NPAD * NPAD + 255) / 256, 256, 0, stream>>>(A, Y, W, cvec);
    // 5) X = P W + c  — the 34.8 GFLOP batch GEMM on the f32 WMMA pipe
    {
        dim3 grid(NB / 256, NPAD / 128);
        k_gemm<<<grid, 256, 0, stream>>>(puzzles, W, cvec, X);
    }
}